// SAGE_16965120819594
// MI455X (gfx1250) — compile-verified
//
#include <hip/hip_runtime.h>

typedef __attribute__((ext_vector_type(2))) float v2f;
typedef __attribute__((ext_vector_type(8))) float v8f;
typedef __attribute__((ext_vector_type(4))) unsigned int v4u;
typedef __attribute__((ext_vector_type(8))) int v8si;
typedef __attribute__((ext_vector_type(4))) int v4si;

#define FDIM 256   // IN == HID == 256
#define KCH  32    // K-chunk staged into LDS per iteration
#define NT   64    // N columns per block
#define MW   8     // waves per block
#define MT   (MW * 16)  // 128 rows per block

#if __has_builtin(__builtin_amdgcn_tensor_load_to_lds)
#define HAVE_TDM 1
#else
#define HAVE_TDM 0
#endif

// ---------------------------------------------------------------------------
// Edge scatter: one wave (32 lanes) per edge. Each lane atomically adds 8
// strided floats (coalesced 128B groups) of x[src] into agg[tgt]; lane 0
// bumps the degree counter. agg/cnt fit in L2, so atomics resolve there.
// ---------------------------------------------------------------------------
__global__ __launch_bounds__(256) void sage_scatter(
    const float* __restrict__ x, const int* __restrict__ src,
    const int* __restrict__ tgt, float* __restrict__ agg,
    float* __restrict__ cnt, int E)
{
    const int wave = (blockIdx.x * blockDim.x + threadIdx.x) >> 5;
    const int lane = threadIdx.x & 31;
    if (wave >= E) return;
    const int s = src[wave];
    const int t = tgt[wave];
    const float* __restrict__ xs = x + (size_t)s * FDIM;
    float* __restrict__ ag = agg + (size_t)t * FDIM;
#pragma unroll
    for (int j = 0; j < FDIM / 32; ++j)
        atomicAdd(&ag[lane + 32 * j], xs[lane + 32 * j]);
    if (lane == 0) atomicAdd(&cnt[t], 1.0f);
}

// ---------------------------------------------------------------------------
// inv[i] = 1 / max(cnt[i], 1)
// ---------------------------------------------------------------------------
__global__ __launch_bounds__(256) void sage_invcnt(
    const float* __restrict__ cnt, float* __restrict__ inv, int n)
{
    const int i = blockIdx.x * blockDim.x + threadIdx.x;
    if (i < n) inv[i] = 1.0f / fmaxf(cnt[i], 1.0f);
}

#if HAVE_TDM
// ---------------------------------------------------------------------------
// TDM: stage a KCH x NT f32 tile (row stride `ld` elements) from global into
// LDS at byte offset ldsOff. D# packed per CDNA5 ISA 8.3/8.4:
//   group0: count=1 | lds_addr | global_addr[56:0] | type=2
//   group1: data_size=4B, tensor_dim0=NT, tensor_dim1=KCH, tile_dim0=NT,
//           tile_dim1=KCH, tensor_dim0_stride=ld
// 2D tile: remaining groups zero ("unused" encoding). This toolchain's
// builtin is the 6-arg clang-23 form (extra int32x8 group before cpol).
// ---------------------------------------------------------------------------
__device__ __forceinline__ void tdm_stage(unsigned ldsOff, const float* gsrc, int ld)
{
    unsigned long long ga = (unsigned long long)(uintptr_t)gsrc;
    v4u g0;
    g0.x = 1u;                                   // count = 1 (valid user D#)
    g0.y = ldsOff;                               // lds_addr (bytes)
    g0.z = (unsigned)ga;                         // global_addr[31:0]
    g0.w = (unsigned)(ga >> 32) | (2u << 30);    // global_addr[56:32] | type=2
    v8si g1;
    g1[0] = 2 << 16;                             // data_size = 2 (4 bytes)
    g1[1] = NT << 16;                            // tensor_dim0[15:0]
    g1[2] = KCH << 16;                           // tensor_dim1[15:0]
    g1[3] = NT << 16;                            // tile_dim0
    g1[4] = KCH;                                 // tile_dim1 (tile_dim2 = 0)
    g1[5] = ld;                                  // tensor_dim0_stride[31:0]
    g1[6] = 0;
    g1[7] = 0;
    v4si z4 = {0, 0, 0, 0};
    v8si z8 = {0, 0, 0, 0, 0, 0, 0, 0};
    __builtin_amdgcn_tensor_load_to_lds(g0, g1, z4, z4, z8, 0);
}
#endif

// ---------------------------------------------------------------------------
// Fused SAGE GEMM, f32 WMMA (V_WMMA_F32_16X16X4_F32):
//   out[r, :] = act( (agg[r]*inv[r]) @ wl + xt[r] @ wr + bias )
// i.e. a K=512 GEMM with A = [agg*inv | xt].
// 8 waves/block; wave w owns rows [blk*128 + w*16, +16), all waves share a
// double-buffered KCH x NT weight chunk in LDS, staged by the Tensor Data
// Mover (wave 0) and fenced with s_wait_tensorcnt + workgroup barrier.
// Fragment layouts per ISA 7.12.2; EXEC all-ones throughout the WMMA region.
// ---------------------------------------------------------------------------
__global__ __launch_bounds__(256) void sage_wmma_gemm(
    const float* __restrict__ agg, const float* __restrict__ inv,
    const float* __restrict__ xt,
    const float* __restrict__ wl, const float* __restrict__ wr,  // [256, OUTD]
    const float* __restrict__ bias,
    float* __restrict__ out, int OUTD, int relu)
{
    __shared__ float sb[2][KCH * NT];

    const int wave  = threadIdx.x >> 5;
    const int lane  = threadIdx.x & 31;
    const int m     = lane & 15;
    const int khalf = (lane >> 4) * 2;       // 0 for lanes 0-15, 2 for 16-31
    const int rowTile = blockIdx.x * MT + wave * 16;
    const int colBase = blockIdx.y * NT;

    const int row = rowTile + m;
    const float invc = inv[row];
    const float* __restrict__ arow = agg + (size_t)row * FDIM;
    const float* __restrict__ xrow = xt  + (size_t)row * FDIM;

    v8f acc[4] = {};

    const int nChunks = (2 * FDIM) / KCH;    // 16: first 8 from wl, next 8 wr

#if HAVE_TDM
    const unsigned ldsBase = (unsigned)(unsigned long long)(uintptr_t)&sb[0][0];
    if (wave == 0)
        tdm_stage(ldsBase, wl + colBase, OUTD);          // chunk 0 -> buf 0
#endif

    for (int c = 0; c < nChunks; ++c) {
        const int half = (c >= nChunks / 2);
        const int krel = (c & 7) * KCH;                  // K offset within half

#if HAVE_TDM
        if (wave == 0) __builtin_amdgcn_s_wait_tensorcnt(0);
        __syncthreads();                                 // buf[c&1] published
        if (c + 1 < nChunks && wave == 0) {
            const int c1 = c + 1;
            const float* wsrc = (c1 < 8
                ? wl + (size_t)(c1 * KCH) * OUTD
                : wr + (size_t)((c1 - 8) * KCH) * OUTD) + colBase;
            tdm_stage(ldsBase + (unsigned)((c1 & 1) * (KCH * NT * 4)), wsrc, OUTD);
        }
#else
        {   // cooperative synchronous staging fallback
            const float* wsrc = (c < 8
                ? wl + (size_t)(c * KCH) * OUTD
                : wr + (size_t)((c - 8) * KCH) * OUTD) + colBase;
            for (int i = threadIdx.x; i < KCH * NT; i += 256)
                sb[c & 1][i] = wsrc[(size_t)(i >> 6) * OUTD + (i & 63)];
            __syncthreads();
        }
#endif

        const float* __restrict__ src = half ? xrow : arow;
        const float  scale = half ? 1.0f : invc;
        const float* __restrict__ sbuf = sb[c & 1];

#pragma unroll
        for (int kl = 0; kl < KCH; kl += 4) {
            const int kk = kl + khalf;
            v2f a;
            a.x = src[krel + kk]     * scale;
            a.y = src[krel + kk + 1] * scale;
#pragma unroll
            for (int t = 0; t < 4; ++t) {
                const int n = t * 16 + m;
                v2f b;
                b.x = sbuf[kk * NT + n];
                b.y = sbuf[(kk + 1) * NT + n];
                acc[t] = __builtin_amdgcn_wmma_f32_16x16x4_f32(
                    false, a, false, b, (short)0, acc[t], false, false);
            }
        }
        __syncthreads();                                 // done reading buf[c&1]
    }

    // Store with fused bias + optional ReLU (C layout: VGPR i -> M=i / M=i+8).
    const int mbase = rowTile + ((lane >> 4) << 3);
#pragma unroll
    for (int t = 0; t < 4; ++t) {
        const int n = colBase + t * 16 + m;
        const float bv = bias[n];
#pragma unroll
        for (int i = 0; i < 8; ++i) {
            float v = acc[t][i] + bv;
            if (relu) v = fmaxf(v, 0.0f);
            out[(size_t)(mbase + i) * OUTD + n] = v;
        }
    }
}

// ---------------------------------------------------------------------------
// Host-side orchestration (graph-capture safe: only async ops on `stream`).
// ---------------------------------------------------------------------------
extern "C" void kernel_launch(void* const* d_in, const int* in_sizes, int n_in,
                              void* d_out, int out_size, void* d_ws, size_t ws_size,
                              hipStream_t stream)
{
    const float* x   = (const float*)d_in[0];
    const int*   ei0 = (const int*)d_in[1];   // [2, E0]: row0 src, row1 tgt
    const int*   ei1 = (const int*)d_in[2];   // [2, E1]
    const float* wl0 = (const float*)d_in[3];
    const float* bl0 = (const float*)d_in[4];
    const float* wr0 = (const float*)d_in[5];
    const float* wl1 = (const float*)d_in[6];
    const float* bl1 = (const float*)d_in[7];
    const float* wr1 = (const float*)d_in[8];
    float* out = (float*)d_out;

    const int E0 = in_sizes[1] / 2;
    const int E1 = in_sizes[2] / 2;
    constexpr int N1 = 11264, N2 = 1024, OUTD = 128;

    // Workspace layout (floats). Prefix [agg0|cnt0|agg1|cnt1] is zeroed once
    // per call; inv0/inv1/h are fully overwritten every call.
    float* ws   = (float*)d_ws;
    float* agg0 = ws;  ws += (size_t)N1 * FDIM;
    float* cnt0 = ws;  ws += N1;
    float* agg1 = ws;  ws += (size_t)N2 * FDIM;
    float* cnt1 = ws;  ws += N2;
    const size_t zeroBytes = (size_t)((char*)ws - (char*)d_ws);
    float* inv0 = ws;  ws += N1;
    float* inv1 = ws;  ws += N2;
    float* h    = ws;  ws += (size_t)N1 * FDIM;

    (void)hipMemsetAsync(d_ws, 0, zeroBytes, stream);

    // hop 0: N0 -> N1, ReLU
    sage_scatter<<<(E0 + 7) / 8, 256, 0, stream>>>(x, ei0, ei0 + E0, agg0, cnt0, E0);
    sage_invcnt<<<(N1 + 255) / 256, 256, 0, stream>>>(cnt0, inv0, N1);
    sage_wmma_gemm<<<dim3(N1 / MT, FDIM / NT), 256, 0, stream>>>(
        agg0, inv0, x, wl0, wr0, bl0, h, FDIM, 1);

    // hop 1: N1 -> N2, no ReLU, writes d_out
    sage_scatter<<<(E1 + 7) / 8, 256, 0, stream>>>(h, ei1, ei1 + E1, agg1, cnt1, E1);
    sage_invcnt<<<(N2 + 255) / 256, 256, 0, stream>>>(cnt1, inv1, N2);
    sage_wmma_gemm<<<dim3(N2 / MT, OUTD / NT), 256, 0, stream>>>(
        agg1, inv1, h, wl1, wr1, bl1, out, OUTD, 0);
}